// MyGNN_35485019799700
// MI455X (gfx1250) — compile-verified
//
#include <hip/hip_runtime.h>
#include <hip/hip_bf16.h>

#define N_NODES 20000
#define N_EDGES 640000
#define D 512

typedef __attribute__((ext_vector_type(2))) float v2f;
typedef __attribute__((ext_vector_type(8))) float v8f;

// ---------------------------------------------------------------- utilities

__global__ void zero_kernel(float* __restrict__ p, int count) {
    int i = blockIdx.x * blockDim.x + threadIdx.x;
    if (i < count) p[i] = 0.0f;
}

// h[n][:] = emb[in_feat[n]][:]
__global__ void embed_kernel(const int* __restrict__ in_feat,
                             const float* __restrict__ emb,
                             float* __restrict__ h) {
    int idx = blockIdx.x * blockDim.x + threadIdx.x;   // over N_NODES * D/4
    int n = idx / (D / 4);
    int c = (idx % (D / 4)) * 4;
    if (n >= N_NODES) return;
    int t = in_feat[n];
    *(float4*)(h + (size_t)n * D + c) =
        *(const float4*)(emb + (size_t)t * D + c);
}

// deg[dst[e]] += 1
__global__ void deg_kernel(const int* __restrict__ dst, float* __restrict__ deg) {
    int e = blockIdx.x * blockDim.x + threadIdx.x;
    if (e >= N_EDGES) return;
    atomicAdd(&deg[dst[e]], 1.0f);
}

// msg[dst[e]][:] += h[src[e]][:]   (one thread per edge x 4-float chunk)
__global__ void scatter_kernel(const float* __restrict__ h,
                               const int* __restrict__ src,
                               const int* __restrict__ dst,
                               float* __restrict__ msg) {
    unsigned idx = blockIdx.x * blockDim.x + threadIdx.x;  // N_EDGES * 128 threads
    unsigned e = idx >> 7;          // D/4 = 128 chunks per edge
    unsigned c = (idx & 127u) * 4u;
    if (e >= N_EDGES) return;
    int s = src[e], d = dst[e];
    float4 v = *(const float4*)(h + (size_t)s * D + c);
    float* p = msg + (size_t)d * D + c;
    atomicAdd(p + 0, v.x);
    atomicAdd(p + 1, v.y);
    atomicAdd(p + 2, v.z);
    atomicAdd(p + 3, v.w);
}

// ------------------------------------------------- fused SAGE GEMM + ReLU
// out = relu( H @ Ws + (MSG/max(deg,1)) @ Wn + bias )
// grid: (N_NODES/16, D/128), block: 256 threads = 8 wave32s.
// Each wave owns a 16x16 output tile; block stages the 16-row H / normalized
// MSG slabs (16x512 each) into LDS once and sweeps K with f32 WMMA (K=4/step).
__global__ __launch_bounds__(256) void sage_gemm_relu_kernel(
    const float* __restrict__ h, const float* __restrict__ msg,
    const float* __restrict__ deg, const float* __restrict__ Ws,
    const float* __restrict__ Wn, const float* __restrict__ bias,
    float* __restrict__ out) {
    __shared__ float lh[16 * D];
    __shared__ float ln[16 * D];

    const int row0 = blockIdx.x * 16;
    const int col0 = blockIdx.y * 128;
    const int tid = threadIdx.x;

    // Stage H tile and degree-normalized MSG tile into LDS (float4 per trip).
    for (int i = tid; i < 16 * D / 4; i += 256) {
        int r = (i * 4) / D;
        int c = (i * 4) % D;
        float4 v = *(const float4*)(h + (size_t)(row0 + r) * D + c);
        *(float4*)(lh + r * D + c) = v;
        float inv = 1.0f / fmaxf(deg[row0 + r], 1.0f);
        float4 m = *(const float4*)(msg + (size_t)(row0 + r) * D + c);
        m.x *= inv; m.y *= inv; m.z *= inv; m.w *= inv;
        *(float4*)(ln + r * D + c) = m;
    }
    __syncthreads();

    const int lane = tid & 31;
    const int wave = tid >> 5;
    const int nloc = lane & 15;          // N within tile
    const int half = lane >> 4;          // half-wave: K split for A/B frags
    const int col  = col0 + wave * 16 + nloc;

    // C/D f32 16x16 layout: VGPR r -> M=r (lanes 0-15) / M=r+8 (lanes 16-31),
    // N = lane&15. Bias is per-column, so broadcast into all 8 accumulators.
    float bv = bias[col];
    v8f acc;
    for (int r = 0; r < 8; ++r) acc[r] = bv;

    // A f32 16x4 layout: lane(M=lane&15) holds {K=k0, K=k0+1} with
    // k0 = k + 2*half.  B 4x16 mirrors it: lane(N) holds {B[k0][N], B[k0+1][N]}.
    for (int k = 0; k < D; k += 4) {
        const int ka = k + half * 2;

        v2f a = *(const v2f*)(lh + nloc * D + ka);            // ds_load_b64
        v2f bs;
        bs.x = Ws[(size_t)ka * D + col];
        bs.y = Ws[(size_t)(ka + 1) * D + col];
        acc = __builtin_amdgcn_wmma_f32_16x16x4_f32(
            false, a, false, bs, (short)0, acc, false, false);

        v2f an = *(const v2f*)(ln + nloc * D + ka);
        v2f bn;
        bn.x = Wn[(size_t)ka * D + col];
        bn.y = Wn[(size_t)(ka + 1) * D + col];
        acc = __builtin_amdgcn_wmma_f32_16x16x4_f32(
            false, an, false, bn, (short)0, acc, false, false);
    }

    const int mbase = row0 + half * 8;
    for (int r = 0; r < 8; ++r) {
        out[(size_t)(mbase + r) * D + col] = fmaxf(acc[r], 0.0f);
    }
}

// ---------------------------------------------------------------- launcher

extern "C" void kernel_launch(void* const* d_in, const int* in_sizes, int n_in,
                              void* d_out, int out_size, void* d_ws, size_t ws_size,
                              hipStream_t stream) {
    const int*   in_feat = (const int*)d_in[0];
    const int*   src     = (const int*)d_in[1];
    const int*   dst     = (const int*)d_in[2];
    const float* emb     = (const float*)d_in[3];
    const float* Ws0     = (const float*)d_in[4];
    const float* Wn0     = (const float*)d_in[5];
    const float* b0      = (const float*)d_in[6];
    const float* Ws1     = (const float*)d_in[7];
    const float* Wn1     = (const float*)d_in[8];
    const float* b1      = (const float*)d_in[9];
    float*       out     = (float*)d_out;

    const size_t feat_elems = (size_t)N_NODES * D;   // 10,240,000 floats
    float* h0  = (float*)d_ws;
    float* msg = h0 + feat_elems;
    float* h1  = msg + feat_elems;
    float* deg = h1 + feat_elems;                    // 20,000 floats

    const int T = 256;
    const int featGrid    = (int)((feat_elems / 4 + T - 1) / T);       // float4 trips
    const int zeroGridMsg = (int)((feat_elems + T - 1) / T);
    const int zeroGridDeg = (N_NODES + T - 1) / T;
    const int edgeGrid    = (N_EDGES + T - 1) / T;
    const int scatGrid    = (int)(((size_t)N_EDGES * (D / 4) + T - 1) / T);
    dim3 gemmGrid(N_NODES / 16, D / 128);

    // h0 = emb[in_feat]
    embed_kernel<<<featGrid, T, 0, stream>>>(in_feat, emb, h0);

    // ---- layer 1 ----
    zero_kernel<<<zeroGridMsg, T, 0, stream>>>(msg, (int)feat_elems);
    zero_kernel<<<zeroGridDeg, T, 0, stream>>>(deg, N_NODES);
    deg_kernel<<<edgeGrid, T, 0, stream>>>(dst, deg);
    scatter_kernel<<<scatGrid, T, 0, stream>>>(h0, src, dst, msg);
    sage_gemm_relu_kernel<<<gemmGrid, T, 0, stream>>>(h0, msg, deg, Ws0, Wn0, b0, h1);

    // ---- layer 2 ----
    zero_kernel<<<zeroGridMsg, T, 0, stream>>>(msg, (int)feat_elems);
    scatter_kernel<<<scatGrid, T, 0, stream>>>(h1, src, dst, msg);
    sage_gemm_relu_kernel<<<gemmGrid, T, 0, stream>>>(h1, msg, deg, Ws1, Wn1, b1, out);
}